// NTMReadHead_62277025792414
// MI455X (gfx1250) — compile-verified
//
#include <hip/hip_runtime.h>
#include <math.h>

#define B_ 64
#define N_ 16384
#define M_ 64
#define D_ 512
#define OC 80        // o columns padded 70 -> 80 (5 WMMA col tiles)
#define NOUT 70
#define EPSF 1e-16f
#define NC 16        // chunks per batch for pass 2
#define CH (N_ / NC) // 1024 rows per chunk

typedef __attribute__((ext_vector_type(2))) float v2f;
typedef __attribute__((ext_vector_type(8))) float v8f;

// ---------------- Kernel 1: o = ctrl @ W^T + bias  via v_wmma_f32_16x16x4_f32
// 20 waves, one 16x16 output tile each; K-loop of 128 x (K=4) WMMA steps.
__global__ __launch_bounds__(640) void k1_proj_wmma(const float* __restrict__ ctrl,
                                                    const float* __restrict__ Wm,
                                                    const float* __restrict__ bias,
                                                    float* __restrict__ o) {
  int tid  = threadIdx.x;
  int wave = tid >> 5;
  int lane = tid & 31;
  int tm = wave / 5;                 // row tile 0..3
  int tn = wave % 5;                 // col tile 0..4
  int mrow  = tm * 16 + (lane & 15); // A row for this lane
  int ncol  = tn * 16 + (lane & 15); // B col / D col for this lane
  int khalf = (lane >> 4) * 2;       // ISA A 16x4 layout: lanes 16-31 hold K+2,K+3
  bool nvalid = (ncol < NOUT);

  const float* arow = ctrl + mrow * D_;
  const float* brow = nvalid ? (Wm + ncol * D_) : Wm; // W row-major (70 x 512): B = W^T col

  v8f c = {};
  for (int k0 = 0; k0 < D_; k0 += 4) {
    v2f a, bv;
    a.x = arow[k0 + khalf];
    a.y = arow[k0 + khalf + 1];
    float b0 = brow[k0 + khalf];
    float b1 = brow[k0 + khalf + 1];
    bv.x = nvalid ? b0 : 0.0f;
    bv.y = nvalid ? b1 : 0.0f;
    c = __builtin_amdgcn_wmma_f32_16x16x4_f32(false, a, false, bv, (short)0, c,
                                              false, false);
  }
  float bb = nvalid ? bias[ncol] : 0.0f;
  int rbase = tm * 16 + ((lane >> 4) ? 8 : 0); // C/D layout: VGPR v -> row v (+8 upper half)
#pragma unroll
  for (int v = 0; v < 8; ++v) {
    o[(rbase + v) * OC + ncol] = c[v] + bb;
  }
}

// ---------------- Kernel 1b: per-batch scalars: beta, g, s(softmax3), gamma, ||k||
// sc layout (floats): [0:64] beta, [64:128] g, [128:192] gamma, [192:256] kn, [256:448] s
__global__ void k1b_act(const float* __restrict__ o, float* __restrict__ sc) {
  int b = threadIdx.x;
  if (b >= B_) return;
  const float* ob = o + b * OC;
  float xb = ob[M_ + 0];
  float beta = fmaxf(xb, 0.0f) + log1pf(expf(-fabsf(xb))); // softplus
  float xg = ob[M_ + 1];
  float g = 1.0f / (1.0f + expf(-xg));
  float s0 = ob[M_ + 2], s1 = ob[M_ + 3], s2 = ob[M_ + 4];
  float mx = fmaxf(s0, fmaxf(s1, s2));
  float e0 = expf(s0 - mx), e1 = expf(s1 - mx), e2 = expf(s2 - mx);
  float es = e0 + e1 + e2;
  float xm = ob[M_ + 5];
  float gamma = 1.0f + fmaxf(xm, 0.0f) + log1pf(expf(-fabsf(xm)));
  float ss = 0.0f;
  for (int m = 0; m < M_; ++m) { float kv = ob[m]; ss += kv * kv; }
  sc[b]        = beta;
  sc[64 + b]   = g;
  sc[128 + b]  = gamma;
  sc[192 + b]  = sqrtf(ss);
  sc[256 + b * 3 + 0] = e0 / es;
  sc[256 + b * 3 + 1] = e1 / es;
  sc[256 + b * 3 + 2] = e2 / es;
}

// ---------------- Kernel 2: big pass 1 — fused dot(k,row) + ||row|| per memory row.
// One row (64 f32 = 256B) per lane, 16 x b128 streaming loads, no cross-lane reduction.
__global__ __launch_bounds__(256) void k2_scores(const float* __restrict__ mem,
                                                 const float* __restrict__ o,
                                                 const float* __restrict__ sc,
                                                 float* __restrict__ a) {
  int tid = threadIdx.x;
  long rid = (long)blockIdx.x * 256 + tid; // global row id in [0, B*N)
  int b = (int)(rid >> 14);                // N = 16384
  __shared__ float kk[M_];
  __shared__ float sb[2];
  if (tid < M_) kk[tid] = o[b * OC + tid]; // k = o[:, :64]
  if (tid == 0) { sb[0] = sc[b]; sb[1] = sc[192 + b]; } // beta, ||k||
  __syncthreads();
  const float4* row = (const float4*)(mem + rid * (long)M_);
  const float4* k4  = (const float4*)kk;
  float num = 0.0f, ss = 0.0f;
#pragma unroll
  for (int i = 0; i < 16; ++i) {
    float4 v  = row[i];
    float4 kv = k4[i];
    num += v.x * kv.x + v.y * kv.y + v.z * kv.z + v.w * kv.w;
    ss  += v.x * v.x  + v.y * v.y  + v.z * v.z  + v.w * v.w;
  }
  float denom = sb[1] * sqrtf(ss) + EPSF;
  a[rid] = sb[0] * num / denom;
}

// ---------------- Kernel 3: row softmax over N + gate with w_prev. One block per batch.
__global__ __launch_bounds__(1024) void k3_softmax_gate(const float* __restrict__ a,
                                                        const float* __restrict__ w_prev,
                                                        const float* __restrict__ sc,
                                                        float* __restrict__ w_g) {
  int b = blockIdx.x;
  int t = threadIdx.x;
  int wid = t >> 5, lane = t & 31;
  const float4* a4  = (const float4*)(a + (long)b * N_);
  const float4* wp4 = (const float4*)(w_prev + (long)b * N_);
  float4*       wg4 = (float4*)(w_g + (long)b * N_);

  __shared__ float red[32];
  __shared__ float bcast;

  float4 x[4];
  float mx = -INFINITY;
#pragma unroll
  for (int i = 0; i < 4; ++i) {
    x[i] = a4[i * 1024 + t];
    mx = fmaxf(mx, fmaxf(fmaxf(x[i].x, x[i].y), fmaxf(x[i].z, x[i].w)));
  }
  for (int off = 16; off; off >>= 1) mx = fmaxf(mx, __shfl_xor(mx, off, 32));
  if (lane == 0) red[wid] = mx;
  __syncthreads();
  if (wid == 0) {
    float v = red[lane];
    for (int off = 16; off; off >>= 1) v = fmaxf(v, __shfl_xor(v, off, 32));
    if (lane == 0) bcast = v;
  }
  __syncthreads();
  mx = bcast;

  float sum = 0.0f;
#pragma unroll
  for (int i = 0; i < 4; ++i) {
    x[i].x = expf(x[i].x - mx); x[i].y = expf(x[i].y - mx);
    x[i].z = expf(x[i].z - mx); x[i].w = expf(x[i].w - mx);
    sum += x[i].x + x[i].y + x[i].z + x[i].w;
  }
  for (int off = 16; off; off >>= 1) sum += __shfl_xor(sum, off, 32);
  if (lane == 0) red[wid] = sum;
  __syncthreads();
  if (wid == 0) {
    float v = red[lane];
    for (int off = 16; off; off >>= 1) v += __shfl_xor(v, off, 32);
    if (lane == 0) bcast = v;
  }
  __syncthreads();
  float inv = 1.0f / bcast;
  float g = sc[64 + b];
#pragma unroll
  for (int i = 0; i < 4; ++i) {
    float4 wp = wp4[i * 1024 + t];
    float4 wv;
    wv.x = g * (x[i].x * inv) + (1.0f - g) * wp.x;
    wv.y = g * (x[i].y * inv) + (1.0f - g) * wp.y;
    wv.z = g * (x[i].z * inv) + (1.0f - g) * wp.z;
    wv.w = g * (x[i].w * inv) + (1.0f - g) * wp.w;
    wg4[i * 1024 + t] = wv;
  }
}

// ---------------- Kernel 4: circular shift blend, ^gamma, normalize -> w (into d_out)
__global__ __launch_bounds__(1024) void k4_shift_pow_norm(const float* __restrict__ w_g,
                                                          const float* __restrict__ sc,
                                                          float* __restrict__ w_out) {
  int b = blockIdx.x;
  int t = threadIdx.x;
  int wid = t >> 5, lane = t & 31;
  float s0 = sc[256 + b * 3 + 0];
  float s1 = sc[256 + b * 3 + 1];
  float s2 = sc[256 + b * 3 + 2];
  float gamma = sc[128 + b];
  const float*  wg  = w_g + (long)b * N_;
  const float4* wg4 = (const float4*)wg;

  __shared__ float red[32];
  __shared__ float bcast;

  float4 p[4];
  float lsum = 0.0f;
#pragma unroll
  for (int i = 0; i < 4; ++i) {
    int idx4 = i * 1024 + t;
    int n0 = idx4 * 4;
    float4 cc = wg4[idx4];
    float lft = wg[(n0 + N_ - 1) & (N_ - 1)];
    float rgt = wg[(n0 + 4) & (N_ - 1)];
    float4 wt;
    wt.x = s0 * lft  + s1 * cc.x + s2 * cc.y;
    wt.y = s0 * cc.x + s1 * cc.y + s2 * cc.z;
    wt.z = s0 * cc.y + s1 * cc.z + s2 * cc.w;
    wt.w = s0 * cc.z + s1 * cc.w + s2 * rgt;
    float4 pw;
    pw.x = powf(wt.x, gamma); pw.y = powf(wt.y, gamma);
    pw.z = powf(wt.z, gamma); pw.w = powf(wt.w, gamma);
    lsum += pw.x + pw.y + pw.z + pw.w;
    p[i] = pw;
  }
  for (int off = 16; off; off >>= 1) lsum += __shfl_xor(lsum, off, 32);
  if (lane == 0) red[wid] = lsum;
  __syncthreads();
  if (wid == 0) {
    float v = red[lane];
    for (int off = 16; off; off >>= 1) v += __shfl_xor(v, off, 32);
    if (lane == 0) bcast = v;
  }
  __syncthreads();
  float inv = 1.0f / (bcast + EPSF);
  float4* wo4 = (float4*)(w_out + (long)b * N_);
#pragma unroll
  for (int i = 0; i < 4; ++i) {
    float4 q;
    q.x = p[i].x * inv; q.y = p[i].y * inv; q.z = p[i].z * inv; q.w = p[i].w * inv;
    wo4[i * 1024 + t] = q;
  }
}

// ---------------- Kernel 5: big pass 2 — per-chunk partial r. Deterministic (no atomics).
// Block = 256 threads = 16 row-groups x 16 float4 slots; fully coalesced b128 streams.
__global__ __launch_bounds__(256) void k5_weighted_partial(const float* __restrict__ mem,
                                                           const float* __restrict__ w,
                                                           float* __restrict__ partial) {
  int b = blockIdx.x / NC;
  int c = blockIdx.x % NC;
  int t = threadIdx.x;
  int grp = t >> 4; // 0..15
  int q   = t & 15; // float4 slot within 64-float row
  const float4* mem4 = (const float4*)mem;
  const float*  wb   = w + (long)b * N_;
  float4 acc = {0.f, 0.f, 0.f, 0.f};
  int nbase = c * CH + grp;
  for (int i = 0; i < CH / 16; ++i) {
    int n = nbase + 16 * i;
    float wn = wb[n];
    float4 v = mem4[((long)b * N_ + n) * 16 + q];
    acc.x += wn * v.x; acc.y += wn * v.y; acc.z += wn * v.z; acc.w += wn * v.w;
  }
  __shared__ float4 lds[256];
  lds[t] = acc;
  __syncthreads();
  for (int off = 8; off; off >>= 1) {
    if (grp < off) {
      float4 o2 = lds[(grp + off) * 16 + q];
      float4 m2 = lds[grp * 16 + q];
      m2.x += o2.x; m2.y += o2.y; m2.z += o2.z; m2.w += o2.w;
      lds[grp * 16 + q] = m2;
    }
    __syncthreads();
  }
  if (grp == 0) {
    float4 r4 = lds[q];
    float* pr = partial + ((long)(b * NC + c)) * M_ + q * 4;
    pr[0] = r4.x; pr[1] = r4.y; pr[2] = r4.z; pr[3] = r4.w;
  }
}

// ---------------- Kernel 6: reduce chunk partials -> r (into d_out)
__global__ __launch_bounds__(256) void k6_reduce_r(const float* __restrict__ partial,
                                                   float* __restrict__ r) {
  int idx = blockIdx.x * 256 + threadIdx.x; // 0..4095
  int b = idx >> 6;
  int m = idx & 63;
  float s = 0.0f;
  for (int c = 0; c < NC; ++c) s += partial[((long)(b * NC + c)) * M_ + m];
  r[idx] = s;
}

// ---------------- launcher
extern "C" void kernel_launch(void* const* d_in, const int* in_sizes, int n_in,
                              void* d_out, int out_size, void* d_ws, size_t ws_size,
                              hipStream_t stream) {
  const float* ctrl   = (const float*)d_in[0]; // (64,512)
  const float* w_prev = (const float*)d_in[1]; // (64,16384)
  const float* mem    = (const float*)d_in[2]; // (64,16384,64)
  const float* Wm     = (const float*)d_in[3]; // (70,512)
  const float* bias   = (const float*)d_in[4]; // (70,)

  float* out_r = (float*)d_out;                 // (64,64)
  float* out_w = (float*)d_out + B_ * M_;       // (64,16384)

  // workspace layout (floats)
  float* ws = (float*)d_ws;
  float* o       = ws;                 // 64*80            = 5120
  float* sc      = ws + 5120;          // 448 (-> 512)
  float* a       = ws + 5632;          // B*N              = 1048576
  float* w_g     = ws + 5632 + 1048576;          // B*N    = 1048576
  float* partial = ws + 5632 + 2 * 1048576;      // B*NC*M = 65536

  k1_proj_wmma<<<1, 640, 0, stream>>>(ctrl, Wm, bias, o);
  k1b_act<<<1, 64, 0, stream>>>(o, sc);
  k2_scores<<<(B_ * N_) / 256, 256, 0, stream>>>(mem, o, sc, a);
  k3_softmax_gate<<<B_, 1024, 0, stream>>>(a, w_prev, sc, w_g);
  k4_shift_pow_norm<<<B_, 1024, 0, stream>>>(w_g, sc, out_w);
  k5_weighted_partial<<<B_ * NC, 256, 0, stream>>>(mem, out_w, partial);
  k6_reduce_r<<<16, 256, 0, stream>>>(partial, out_r);
}